// Shift_7292854469289
// MI455X (gfx1250) — compile-verified
//
#include <hip/hip_runtime.h>
#include <cstdint>

// Shift gather: out[b,c,h,w] = x[b,c, sh[h], sw[w]]
// B=16, C=64, H=W=256, fp32. Pure bandwidth problem (512 MB traffic,
// ~22us floor at 23.3 TB/s). Strategy: precompute index tables, stage the
// W-index table in LDS via the gfx1250 async global->LDS path, and stream
// rows with b128 loads + non-temporal b128 stores (fast path when the
// W-shift is identity, which it is for xpos ~ 1e-8).

typedef float v4f __attribute__((ext_vector_type(4)));

#define SHIFT_H 256
#define SHIFT_W 256

// ---------------- Kernel 1: index tables -------------------------------
__global__ __launch_bounds__(256) void shift_idx_kernel(
    const float* __restrict__ xpos, const float* __restrict__ ypos,
    const int* __restrict__ stride_p, int* __restrict__ sh,
    int* __restrict__ sw) {
  const int t = threadIdx.x;  // 0..255 covers both H and W tables
  const float s = (float)stride_p[0];

  // (int)(float) in C truncates toward zero == Python int() == jnp.trunc
  int dy = (int)(ypos[t] * s);
  int v = t + dy;
  v = v < 0 ? 0 : (v > SHIFT_H - 1 ? SHIFT_H - 1 : v);
  sh[t] = v;

  int dx = (int)(xpos[t] * s);
  int u = t + dx;
  u = u < 0 ? 0 : (u > SHIFT_W - 1 ? SHIFT_W - 1 : u);
  sw[t] = u;
}

// ---------------- Kernel 2: streaming row gather -----------------------
// Grid: (B*C*H / ROWS_PER_BLOCK) blocks of 256 threads.
// Each thread owns one 4-float group of a row (64 groups per 256-wide row),
// and walks 4 rows per block-iteration (256 threads = 4 rows x 64 groups).
__global__ __launch_bounds__(256) void shift_gather_kernel(
    const float* __restrict__ x, const int* __restrict__ sh,
    const int* __restrict__ sw, float* __restrict__ out, int rows_per_block) {
  __shared__ int s_sw[SHIFT_W];

  const int tid = threadIdx.x;

  // Stage sw[] into LDS with the CDNA5 async copy (tracked by ASYNCcnt):
  // each lane copies one dword global -> LDS without a VGPR round trip.
  {
    unsigned lds_off = (unsigned)(uintptr_t)(&s_sw[tid]);  // LDS byte offset
    const int* gsrc = sw + tid;
    asm volatile("global_load_async_to_lds_b32 %0, %1, off"
                 :
                 : "v"(lds_off), "v"(gsrc)
                 : "memory");
    asm volatile("s_wait_asynccnt 0" ::: "memory");
  }
  __syncthreads();

  const int lr0 = tid >> 6;         // local row 0..3
  const int w4 = (tid & 63) << 2;   // group start column

  const int i0 = s_sw[w4 + 0];
  const int i1 = s_sw[w4 + 1];
  const int i2 = s_sw[w4 + 2];
  const int i3 = s_sw[w4 + 3];
  const bool ident =
      (i0 == w4) & (i1 == w4 + 1) & (i2 == w4 + 2) & (i3 == w4 + 3);

  const long rowBase = (long)blockIdx.x * rows_per_block;

  for (int r = lr0; r < rows_per_block; r += 4) {
    const long row = rowBase + r;          // flattened (b*C + c)*H + h
    const int h = (int)(row & (SHIFT_H - 1));
    const long bc = row >> 8;              // H == 256
    const int srow = sh[h];                // wave-uniform, L2-hot

    const float* src = x + (((bc << 8) + srow) << 8);  // (bc*H + srow)*W

    v4f v;
    if (ident) {
      // Contiguous fast path: 128-bit load (RT: keep src rows in L2 since
      // ypos in [-3,3] makes nearby output rows re-read the same src row).
      v = *(const v4f*)(src + w4);
    } else {
      // General W-gather fallback.
      v.x = src[i0];
      v.y = src[i1];
      v.z = src[i2];
      v.w = src[i3];
    }

    // Output is written once and never re-read: non-temporal 128-bit store
    // keeps 256 MB of write traffic out of the 192 MB L2.
    __builtin_nontemporal_store(v, (v4f*)(out + (row << 8) + w4));
  }
}

// ---------------- Launch ------------------------------------------------
extern "C" void kernel_launch(void* const* d_in, const int* in_sizes, int n_in,
                              void* d_out, int out_size, void* d_ws,
                              size_t ws_size, hipStream_t stream) {
  const float* x = (const float*)d_in[0];     // (16,64,256,256) f32
  const float* xpos = (const float*)d_in[1];  // (256,) f32
  const float* ypos = (const float*)d_in[2];  // (256,) f32
  const int* stride = (const int*)d_in[3];    // scalar int

  float* out = (float*)d_out;

  int* sh = (int*)d_ws;        // 256 ints
  int* sw = sh + SHIFT_H;      // 256 ints (2 KB total in d_ws)

  shift_idx_kernel<<<1, 256, 0, stream>>>(xpos, ypos, stride, sh, sw);

  const int ROWS_PER_BLOCK = 16;
  const int totalRows = out_size / SHIFT_W;          // B*C*H = 262144
  const int blocks = totalRows / ROWS_PER_BLOCK;     // 16384
  shift_gather_kernel<<<blocks, 256, 0, stream>>>(x, sh, sw, out,
                                                  ROWS_PER_BLOCK);
}